// ComGNNBank_67293547593780
// MI455X (gfx1250) — compile-verified
//
#include <hip/hip_runtime.h>
#include <hip/hip_bf16.h>

// ---------------------------------------------------------------------------
// ComGNN bank for MI455X (gfx1250, wave32).
//
// Roofline: ~18 GFLOP of GEMM vs ~4 GB of irregular edge gather/scatter
// traffic -> memory/atomic bound (~0.3 ms @ 23.3 TB/s HBM, atomics resolving
// in the 192 MB L2). Strategy:
//   * edge-parallel scatter with global_atomic_add_f32
//   * all GEMMs on the matrix pipe via V_WMMA_F32_16X16X4_F32 (f32 numerics,
//     matching the reference), 5-way M-blocking so each B fragment feeds
//     5 WMMAs (3125 M-tiles = 5 x 625 exactly, no bounds checks)
//   * BN statistics fused into the z-producing GEMM epilogue: LDS ds_add_f32
//     partial reduction + 128 global atomics per block (saves a full 12.8 MB
//     re-read of z per layer per community)
// ---------------------------------------------------------------------------

#define N_NODES 50000
#define N_EDGES 800000
#define IN_DIM  256
#define EMB_DIM 256
#define N_COMS  4
#define D_COM   64
#define BN_EPS  1e-5f
#define MBLK    5          // M-tiles per wave; 3125 = 5 * 625

typedef float v2f __attribute__((ext_vector_type(2)));
typedef float v8f __attribute__((ext_vector_type(8)));

static __device__ __forceinline__ v8f wmma_f32_k4(v2f a, v2f b, v8f c) {
    // 8 args: (neg_a, A, neg_b, B, c_mod, C, reuse_a, reuse_b)
    return __builtin_amdgcn_wmma_f32_16x16x4_f32(
        false, a, false, b, (short)0, c, false, false);
}

// ---------------------------------------------------------------------------
// enc = x @ encW + encb          x:[N,256]  encW:[256,256]  out:[N,256]
// Wave computes MBLK 16x16 tiles down one N-tile column (B-fragment reuse x5).
// grid (625, 2) x block 256 (8 waves -> 16 N-tiles).
// ---------------------------------------------------------------------------
__global__ void enc_gemm_kernel(const float* __restrict__ x,
                                const float* __restrict__ W,
                                const float* __restrict__ bias,
                                float* __restrict__ out) {
    const int lane   = threadIdx.x & 31;
    const int wave   = threadIdx.x >> 5;            // 0..7
    const int m_base = blockIdx.x * (16 * MBLK);    // 625 blocks
    const int n0     = (blockIdx.y * 8 + wave) * 16;
    const int l15    = lane & 15;
    const int half   = lane >> 4;                   // 0 or 1

    const float* bcol = W + n0 + l15;

    v8f c[MBLK];
#pragma unroll
    for (int t = 0; t < MBLK; ++t) { v8f zv = {}; c[t] = zv; }

    for (int k0 = 0; k0 < IN_DIM; k0 += 4) {
        const int kk = k0 + half * 2;
        v2f b; b.x = bcol[kk * EMB_DIM]; b.y = bcol[(kk + 1) * EMB_DIM];
#pragma unroll
        for (int t = 0; t < MBLK; ++t) {
            const float* arow = x + (size_t)(m_base + t * 16 + l15) * IN_DIM;
            v2f a; a.x = arow[kk]; a.y = arow[kk + 1];
            c[t] = wmma_f32_k4(a, b, c[t]);
        }
    }

    const int   col = n0 + l15;
    const float bv  = bias[col];
#pragma unroll
    for (int t = 0; t < MBLK; ++t) {
#pragma unroll
        for (int v = 0; v < 8; ++v) {
            const int row = m_base + t * 16 + half * 8 + v;
            out[(size_t)row * EMB_DIM + col] = c[t][v] + bv;
        }
    }
}

// ---------------------------------------------------------------------------
// Weighted scatter-add aggregation: agg[dst[e]] += ew[e] * h[src[e]]
// Thread = (edge, 4-feature group). h may be a strided column block.
// ---------------------------------------------------------------------------
__global__ void agg_kernel(const int* __restrict__ src,
                           const int* __restrict__ dst,
                           const float* __restrict__ ew,
                           const float* __restrict__ h, int hstride, int F,
                           float* __restrict__ agg, int total) {
    const int idx = blockIdx.x * blockDim.x + threadIdx.x;
    if (idx >= total) return;
    const int groups = F >> 2;
    const int e = idx / groups;
    const int g = idx - e * groups;

    const int   s = src[e];
    const int   d = dst[e];
    const float w = ew[e];

    const float4 hv = *(const float4*)(h + (size_t)s * hstride + g * 4);
    float* ap = agg + (size_t)d * F + g * 4;
    atomicAdd(ap + 0, w * hv.x);
    atomicAdd(ap + 1, w * hv.y);
    atomicAdd(ap + 2, w * hv.z);
    atomicAdd(ap + 3, w * hv.w);
}

// ---------------------------------------------------------------------------
// out[N,64] = act( (A (+ A2)) @ W + bias )     W:[K,64]
// 4 waves/block cover the 64 output columns; each wave does MBLK M-tiles.
// stats != nullptr: accumulate per-column sum / sum^2 (LDS reduce + 128
// global atomics per block) for the following BatchNorm.
// ---------------------------------------------------------------------------
__global__ void mlp_gemm_kernel(const float* __restrict__ A,  int strideA,
                                const float* __restrict__ A2, int strideA2,
                                const float* __restrict__ W,
                                const float* __restrict__ bias,
                                float* __restrict__ out,
                                int K, int relu_out,
                                float* __restrict__ stats) {
    __shared__ float ssum[D_COM];
    __shared__ float ssq[D_COM];

    const int lane   = threadIdx.x & 31;
    const int wave   = threadIdx.x >> 5;            // 0..3
    const int m_base = blockIdx.x * (16 * MBLK);    // 625 blocks
    const int n0     = wave * 16;
    const int l15    = lane & 15;
    const int half   = lane >> 4;

    if (stats) {
        if (threadIdx.x < D_COM) { ssum[threadIdx.x] = 0.f; ssq[threadIdx.x] = 0.f; }
        __syncthreads();
    }

    const float* bcol = W + n0 + l15;

    v8f c[MBLK];
#pragma unroll
    for (int t = 0; t < MBLK; ++t) { v8f zv = {}; c[t] = zv; }

    for (int k0 = 0; k0 < K; k0 += 4) {
        const int kk = k0 + half * 2;
        v2f b; b.x = bcol[kk * D_COM]; b.y = bcol[(kk + 1) * D_COM];
#pragma unroll
        for (int t = 0; t < MBLK; ++t) {
            const int m0 = m_base + t * 16;
            const float* arow = A + (size_t)(m0 + l15) * strideA;
            v2f a;
            a.x = arow[kk];
            a.y = arow[kk + 1];
            if (A2) {
                const float* a2 = A2 + (size_t)(m0 + l15) * strideA2;
                a.x += a2[kk];
                a.y += a2[kk + 1];
            }
            c[t] = wmma_f32_k4(a, b, c[t]);
        }
    }

    const int   col = n0 + l15;
    const float bv  = bias[col];
#pragma unroll
    for (int t = 0; t < MBLK; ++t) {
        float lsum = 0.f, lsq = 0.f;
#pragma unroll
        for (int v = 0; v < 8; ++v) {
            const int row = m_base + t * 16 + half * 8 + v;
            float val = c[t][v] + bv;
            if (relu_out) val = fmaxf(val, 0.0f);
            out[(size_t)row * D_COM + col] = val;
            lsum += val;
            lsq  += val * val;
        }
        if (stats) {
            atomicAdd(&ssum[col], lsum);   // ds_add_f32, 64-bank friendly
            atomicAdd(&ssq[col],  lsq);
        }
    }

    if (stats) {
        __syncthreads();
        if (threadIdx.x < D_COM) {
            atomicAdd(&stats[threadIdx.x],         ssum[threadIdx.x]);
            atomicAdd(&stats[D_COM + threadIdx.x], ssq[threadIdx.x]);
        }
    }
}

// ---------------------------------------------------------------------------
// h = relu(gamma*(z-mu)*rsqrt(var+eps)+beta) written into a strided column
// block of the output tensor.
// ---------------------------------------------------------------------------
__global__ void bn_relu_kernel(const float* __restrict__ z,
                               const float* __restrict__ stats,
                               const float* __restrict__ gamma,
                               const float* __restrict__ beta,
                               float* __restrict__ out, int out_stride, int n) {
    const int i = blockIdx.x * blockDim.x + threadIdx.x;
    if (i >= n) return;
    const int row = i >> 6;
    const int col = i & (D_COM - 1);
    const float inv_n = 1.0f / (float)N_NODES;
    const float mean  = stats[col] * inv_n;
    const float var   = stats[D_COM + col] * inv_n - mean * mean;
    const float v = gamma[col] * (z[i] - mean) * rsqrtf(var + BN_EPS) + beta[col];
    out[(size_t)row * out_stride + col] = fmaxf(v, 0.0f);
}

// ---------------------------------------------------------------------------
// Host-side orchestration (graph-capture safe: only async launches/memsets).
// ---------------------------------------------------------------------------
extern "C" void kernel_launch(void* const* d_in, const int* in_sizes, int n_in,
                              void* d_out, int out_size, void* d_ws, size_t ws_size,
                              hipStream_t stream) {
    const float* x    = (const float*)d_in[0];
    const int*   ei   = (const int*)  d_in[1];
    const float* ew   = (const float*)d_in[2];
    const float* W1_0 = (const float*)d_in[3];
    const float* b1_0 = (const float*)d_in[4];
    const float* W2_0 = (const float*)d_in[5];
    const float* b2_0 = (const float*)d_in[6];
    const float* g0   = (const float*)d_in[7];
    const float* bt0  = (const float*)d_in[8];
    const float* W1_1 = (const float*)d_in[9];
    const float* b1_1 = (const float*)d_in[10];
    const float* W2_1 = (const float*)d_in[11];
    const float* b2_1 = (const float*)d_in[12];
    const float* g1   = (const float*)d_in[13];
    const float* bt1  = (const float*)d_in[14];
    const float* encW = (const float*)d_in[15];
    const float* encb = (const float*)d_in[16];

    const int* src = ei;
    const int* dst = ei + N_EDGES;

    float* out_enc = (float*)d_out;                              // [N,256]
    float* out1    = out_enc + (size_t)N_NODES * EMB_DIM;        // [N,256]
    float* out2    = out1    + (size_t)N_NODES * EMB_DIM;        // [N,256]

    // Workspace partition (floats): agg 12.8M | t 3.2M | z 3.2M | stats 128
    float* agg   = (float*)d_ws;
    float* t     = agg + (size_t)N_NODES * IN_DIM;
    float* z     = t   + (size_t)N_NODES * D_COM;
    float* stats = z   + (size_t)N_NODES * D_COM;

    const int MBLOCKS = N_NODES / (16 * MBLK);   // 625 exactly

    // ---- input encoder: one WMMA GEMM ----
    enc_gemm_kernel<<<dim3(MBLOCKS, 2), 256, 0, stream>>>(x, encW, encb, out_enc);

    for (int k = 0; k < N_COMS; ++k) {
        const float* ewk = ew + (size_t)k * N_EDGES;

        // ================= layer 0 (in: x [N,256]) =================
        hipMemsetAsync(agg, 0, (size_t)N_NODES * IN_DIM * sizeof(float), stream);
        {
            const int total  = N_EDGES * (IN_DIM / 4);
            const int blocks = (total + 255) / 256;
            agg_kernel<<<blocks, 256, 0, stream>>>(src, dst, ewk,
                                                   x, IN_DIM, IN_DIM, agg, total);
        }
        // t = relu((x + agg) @ W1_0[k] + b1_0[k])
        mlp_gemm_kernel<<<MBLOCKS, 128, 0, stream>>>(
            x, IN_DIM, agg, IN_DIM,
            W1_0 + (size_t)k * IN_DIM * D_COM, b1_0 + (size_t)k * D_COM,
            t, IN_DIM, 1, nullptr);
        // z = t @ W2_0[k] + b2_0[k]   (+ fused BN statistics)
        hipMemsetAsync(stats, 0, 2 * D_COM * sizeof(float), stream);
        mlp_gemm_kernel<<<MBLOCKS, 128, 0, stream>>>(
            t, D_COM, nullptr, 0,
            W2_0 + (size_t)k * D_COM * D_COM, b2_0 + (size_t)k * D_COM,
            z, D_COM, 0, stats);
        {
            const int n = N_NODES * D_COM;
            bn_relu_kernel<<<(n + 255) / 256, 256, 0, stream>>>(
                z, stats, g0 + (size_t)k * D_COM, bt0 + (size_t)k * D_COM,
                out1 + (size_t)k * D_COM, EMB_DIM, n);
        }

        // ============ layer 1 (in: h1_k = out1[:, k*64:(k+1)*64]) ============
        const float* h1 = out1 + (size_t)k * D_COM;   // row stride EMB_DIM
        hipMemsetAsync(agg, 0, (size_t)N_NODES * D_COM * sizeof(float), stream);
        {
            const int total  = N_EDGES * (D_COM / 4);
            const int blocks = (total + 255) / 256;
            agg_kernel<<<blocks, 256, 0, stream>>>(src, dst, ewk,
                                                   h1, EMB_DIM, D_COM, agg, total);
        }
        mlp_gemm_kernel<<<MBLOCKS, 128, 0, stream>>>(
            h1, EMB_DIM, agg, D_COM,
            W1_1 + (size_t)k * D_COM * D_COM, b1_1 + (size_t)k * D_COM,
            t, D_COM, 1, nullptr);
        hipMemsetAsync(stats, 0, 2 * D_COM * sizeof(float), stream);
        mlp_gemm_kernel<<<MBLOCKS, 128, 0, stream>>>(
            t, D_COM, nullptr, 0,
            W2_1 + (size_t)k * D_COM * D_COM, b2_1 + (size_t)k * D_COM,
            z, D_COM, 0, stats);
        {
            const int n = N_NODES * D_COM;
            bn_relu_kernel<<<(n + 255) / 256, 256, 0, stream>>>(
                z, stats, g1 + (size_t)k * D_COM, bt1 + (size_t)k * D_COM,
                out2 + (size_t)k * D_COM, EMB_DIM, n);
        }
    }
}